// Loss_31516470018602
// MI455X (gfx1250) — compile-verified
//
#include <hip/hip_runtime.h>
#include <math.h>

#define N 2048
#define DIM 768
#define KCL 16
#define ALPHA_C 2.0f
#define MARGIN_C 0.05f
#define EPS_C 1e-6f

__device__ __constant__ float INV_SQRT_D = 0.03608439182435161f;  // 1/sqrt(768)

typedef float v2f __attribute__((ext_vector_type(2)));
typedef float v8f __attribute__((ext_vector_type(8)));
typedef unsigned int v4u __attribute__((ext_vector_type(4)));
typedef int v8i __attribute__((ext_vector_type(8)));
typedef int v4i __attribute__((ext_vector_type(4)));

#if defined(__has_builtin)
#if __has_builtin(__builtin_amdgcn_tensor_load_to_lds) && __has_builtin(__builtin_amdgcn_s_wait_tensorcnt)
#define HAVE_TDM 1
#endif
#endif
#ifndef HAVE_TDM
#define HAVE_TDM 0
#endif

// GEMM tiling
#define BM 128          // workgroup tile rows
#define BN 64           // workgroup tile cols
#define KC 32           // K chunk staged in LDS
#define LDS_STRIDE 35   // 32 data floats + 3 pad floats (TDM pad) -> conflict-free
#define NCHUNK (DIM / KC)

// LDS buffer offsets (floats); computed arithmetically — no pointer arrays
#define A_OFF(b) ((b) * (BM * LDS_STRIDE))
#define B_OFF(b) (2 * BM * LDS_STRIDE + (b) * (BN * LDS_STRIDE))

// ---------------------------------------------------------------------------
// 1) per-row sum and sum-of-squares
// ---------------------------------------------------------------------------
__global__ void row_stats_kernel(const float* __restrict__ x,
                                 float* __restrict__ sq, float* __restrict__ s) {
    __shared__ float ssq[256];
    __shared__ float ssum[256];
    const int row = blockIdx.x;
    const int tid = threadIdx.x;
    const float* xr = x + (size_t)row * DIM;
    float aq = 0.f, as = 0.f;
    for (int i = tid; i < DIM; i += 256) {
        float v = xr[i];
        aq += v * v;
        as += v;
    }
    ssq[tid] = aq;
    ssum[tid] = as;
    __syncthreads();
    for (int off = 128; off > 0; off >>= 1) {
        if (tid < off) {
            ssq[tid]  += ssq[tid + off];
            ssum[tid] += ssum[tid + off];
        }
        __syncthreads();
    }
    if (tid == 0) { sq[row] = ssq[0]; s[row] = ssum[0]; }
}

// ---------------------------------------------------------------------------
// TDM: issue a 2D tile load (rows x KC f32) from x into LDS, padded rows.
// D# per CDNA5 ISA ch.8: group0 = {flags, lds_addr, gaddr_lo, gaddr_hi|type},
// group1 packs data_size/pad + tensor dims/strides + tile dims.
// ---------------------------------------------------------------------------
#if HAVE_TDM
__device__ __forceinline__ void tdm_load_tile(const float* gsrc, unsigned ldsAddr, int rows) {
    unsigned long long ga = (unsigned long long)gsrc;
    v4u g0;
    g0[0] = 1u;                                        // count=1 (valid), user mode
    g0[1] = ldsAddr;                                   // LDS byte address
    g0[2] = (unsigned)(ga & 0xFFFFFFFFu);              // global_addr[31:0]
    g0[3] = (unsigned)((ga >> 32) & 0x01FFFFFFu) | (2u << 30);  // addr[56:32] | type=2
    v8i g1;
    // data_size=4B(2), pad_enable, pad_interval=32 DW (code 4), pad_amount=3 DW (code 2)
    g1[0] = (2 << 16) | (1 << 20) | (4 << 22) | (2 << 25);
    g1[1] = (int)((DIM & 0xFFFF) << 16);               // tensor_dim0[15:0] << 16
    g1[2] = (int)((DIM >> 16) | ((N & 0xFFFF) << 16)); // tensor_dim0[31:16] | tensor_dim1[15:0]
    g1[3] = (int)((N >> 16) | (KC << 16));             // tensor_dim1[31:16] | tile_dim0=KC
    g1[4] = rows;                                      // tile_dim1=rows, tile_dim2=0
    g1[5] = DIM;                                       // tensor_dim0_stride[31:0]
    g1[6] = 0;                                         // stride0 hi | stride1 lo
    g1[7] = 0;
    v4i z4 = {0, 0, 0, 0};
    v8i z8 = {0, 0, 0, 0, 0, 0, 0, 0};
    // clang-23 / therock form: 6 args (g0, g1, g2, g3, g4, cpol)
    __builtin_amdgcn_tensor_load_to_lds(g0, g1, z4, z4, z8, 0);
}
#endif

// ---------------------------------------------------------------------------
// 2) gram = x @ x^T : 8-wave workgroup computes 128x64; wave computes 32x32
//    (2x2 WMMA f32 16x16x4 subtiles); A/B panels staged in LDS by the TDM,
//    double buffered on TENSORcnt.
// ---------------------------------------------------------------------------
__global__ void __launch_bounds__(256)
gram_wmma_kernel(const float* __restrict__ x, float* __restrict__ gram) {
    __shared__ float lds[2 * (BM + BN) * LDS_STRIDE];

    const int bc = blockIdx.x & 31;        // 32 col blocks of 64
    const int br = blockIdx.x >> 5;        // 16 row blocks of 128
    const int rowBase = br * BM;
    const int colBase = bc * BN;

    const int tid  = threadIdx.x;
    const int wid  = tid >> 5;             // wave 0..7
    const int lane = tid & 31;
    const int mn   = lane & 15;
    const int half = lane >> 4;
    const int kh   = half << 1;            // K pair {0,1} vs {2,3}
    const int wr   = (wid >> 1) << 5;      // wave row offset in tile (0..96)
    const int wc   = (wid & 1) << 5;       // wave col offset in tile (0 or 32)

    v8f acc00 = {}, acc01 = {}, acc10 = {}, acc11 = {};

#if HAVE_TDM
    if (wid == 0) {
        tdm_load_tile(x + (size_t)rowBase * DIM,
                      (unsigned)(unsigned long long)(void*)(lds + A_OFF(0)), BM);
        tdm_load_tile(x + (size_t)colBase * DIM,
                      (unsigned)(unsigned long long)(void*)(lds + B_OFF(0)), BN);
    }
    for (int c = 0; c < NCHUNK; ++c) {
        const int cur = c & 1;
        if (wid == 0) {
            if (c + 1 < NCHUNK) {
                const int kc = (c + 1) * KC;
                const int nxt = (c + 1) & 1;
                tdm_load_tile(x + (size_t)rowBase * DIM + kc,
                              (unsigned)(unsigned long long)(void*)(lds + A_OFF(nxt)), BM);
                tdm_load_tile(x + (size_t)colBase * DIM + kc,
                              (unsigned)(unsigned long long)(void*)(lds + B_OFF(nxt)), BN);
                __builtin_amdgcn_s_wait_tensorcnt(2);   // chunk c complete (in-order)
            } else {
                __builtin_amdgcn_s_wait_tensorcnt(0);
            }
        }
        __syncthreads();
        const float* Abase = lds + A_OFF(cur) + (wr + mn) * LDS_STRIDE + kh;
        const float* Bbase = lds + B_OFF(cur) + (wc + mn) * LDS_STRIDE + kh;
#pragma unroll
        for (int kk = 0; kk < KC; kk += 4) {
            v2f a0 = *(const v2f*)(Abase + kk);
            v2f a1 = *(const v2f*)(Abase + 16 * LDS_STRIDE + kk);
            v2f b0 = *(const v2f*)(Bbase + kk);
            v2f b1 = *(const v2f*)(Bbase + 16 * LDS_STRIDE + kk);
            acc00 = __builtin_amdgcn_wmma_f32_16x16x4_f32(false, a0, false, b0, (short)0, acc00, false, false);
            acc01 = __builtin_amdgcn_wmma_f32_16x16x4_f32(false, a0, false, b1, (short)0, acc01, false, false);
            acc10 = __builtin_amdgcn_wmma_f32_16x16x4_f32(false, a1, false, b0, (short)0, acc10, false, false);
            acc11 = __builtin_amdgcn_wmma_f32_16x16x4_f32(false, a1, false, b1, (short)0, acc11, false, false);
        }
        __syncthreads();   // done reading buf before it is overwritten
    }
#else
    for (int c = 0; c < NCHUNK; ++c) {
        const int kc = c * KC;
        for (int idx = tid; idx < BM * KC; idx += 256) {
            int r = idx >> 5, cc = idx & 31;
            lds[A_OFF(0) + r * LDS_STRIDE + cc] = x[(size_t)(rowBase + r) * DIM + kc + cc];
        }
        for (int idx = tid; idx < BN * KC; idx += 256) {
            int r = idx >> 5, cc = idx & 31;
            lds[B_OFF(0) + r * LDS_STRIDE + cc] = x[(size_t)(colBase + r) * DIM + kc + cc];
        }
        __syncthreads();
        const float* Abase = lds + A_OFF(0) + (wr + mn) * LDS_STRIDE + kh;
        const float* Bbase = lds + B_OFF(0) + (wc + mn) * LDS_STRIDE + kh;
#pragma unroll
        for (int kk = 0; kk < KC; kk += 4) {
            v2f a0 = *(const v2f*)(Abase + kk);
            v2f a1 = *(const v2f*)(Abase + 16 * LDS_STRIDE + kk);
            v2f b0 = *(const v2f*)(Bbase + kk);
            v2f b1 = *(const v2f*)(Bbase + 16 * LDS_STRIDE + kk);
            acc00 = __builtin_amdgcn_wmma_f32_16x16x4_f32(false, a0, false, b0, (short)0, acc00, false, false);
            acc01 = __builtin_amdgcn_wmma_f32_16x16x4_f32(false, a0, false, b1, (short)0, acc01, false, false);
            acc10 = __builtin_amdgcn_wmma_f32_16x16x4_f32(false, a1, false, b0, (short)0, acc10, false, false);
            acc11 = __builtin_amdgcn_wmma_f32_16x16x4_f32(false, a1, false, b1, (short)0, acc11, false, false);
        }
        __syncthreads();
    }
#endif

    // C/D layout: VGPR r -> row (r + 8*laneHalf), col = lane&15
    const int grow = rowBase + wr;
    const int gcol = colBase + wc + mn;
#pragma unroll
    for (int r = 0; r < 8; ++r) {
        const int mr = r + (half << 3);
        gram[(size_t)(grow + mr) * N + gcol]           = acc00[r];
        gram[(size_t)(grow + mr) * N + gcol + 16]      = acc01[r];
        gram[(size_t)(grow + 16 + mr) * N + gcol]      = acc10[r];
        gram[(size_t)(grow + 16 + mr) * N + gcol + 16] = acc11[r];
    }
}

// ---------------------------------------------------------------------------
// 3) per-cluster counts, first-negative anchor; zero hinge accumulators
// ---------------------------------------------------------------------------
__global__ void cluster_stats_kernel(const int* __restrict__ y,
                                     const int* __restrict__ labels,
                                     int* __restrict__ lp, int* __restrict__ ln,
                                     int* __restrict__ tt, float* __restrict__ hsum) {
    const int c = threadIdx.x;
    if (c >= KCL) return;
    int plp = 0, pln = 0, t = 0;
    bool found = false;
    for (int i = 0; i < N; ++i) {
        if (labels[i] == c) {
            if (y[i] == 1) {
                ++plp;
            } else {
                ++pln;
                if (!found) { t = i; found = true; }
            }
        }
    }
    lp[c] = plp;
    ln[c] = pln;
    tt[c] = t;       // argmax of bool mask = first True, 0 if none
    hsum[c] = 0.f;   // re-zero every launch (harness does not re-poison)
}

// ---------------------------------------------------------------------------
// 4) d_pn[c,i] = D[i, t_c]
// ---------------------------------------------------------------------------
__global__ void dpn_kernel(const float* __restrict__ gram,
                           const float* __restrict__ sq, const float* __restrict__ s,
                           const int* __restrict__ tt, float* __restrict__ dpn) {
    const int idx = blockIdx.x * 256 + threadIdx.x;   // c*N + i
    if (idx >= KCL * N) return;
    const int c = idx >> 11;
    const int i = idx & (N - 1);
    const int t = tt[c];
    float d2 = sq[i] + sq[t] - 2.f * gram[(size_t)i * N + t]
             + 2.f * EPS_C * (s[i] - s[t]) + (float)DIM * EPS_C * EPS_C;
    dpn[idx] = sqrtf(fmaxf(d2, 0.f)) * INV_SQRT_D;
}

// ---------------------------------------------------------------------------
// 5) hinge sums per cluster over pos-pos pairs
// ---------------------------------------------------------------------------
__global__ void hinge_kernel(const float* __restrict__ gram,
                             const float* __restrict__ sq, const float* __restrict__ s,
                             const int* __restrict__ y, const int* __restrict__ labels,
                             const float* __restrict__ dpn,
                             float* __restrict__ hsum) {
    __shared__ float acc[KCL];
    const int tid = threadIdx.x;
    if (tid < KCL) acc[tid] = 0.f;
    __syncthreads();

    const unsigned idx = blockIdx.x * 256u + (unsigned)tid;  // i*N + j
    const int i = (int)(idx >> 11);
    const int j = (int)(idx & (N - 1));
    const int li = labels[i];
    if (y[i] == 1 && y[j] == 1 && labels[j] == li) {
        float d2 = sq[i] + sq[j] - 2.f * gram[idx]
                 + 2.f * EPS_C * (s[i] - s[j]) + (float)DIM * EPS_C * EPS_C;
        float Dij = sqrtf(fmaxf(d2, 0.f)) * INV_SQRT_D;
        float h = Dij + MARGIN_C - dpn[li * N + i];
        if (h > 0.f) atomicAdd(&acc[li], h);
    }
    __syncthreads();
    if (tid < KCL) {
        float v = acc[tid];
        if (v != 0.f) atomicAdd(&hsum[tid], v);
    }
}

// ---------------------------------------------------------------------------
// 6) class loss (log-softmax) + assemble final scalar
// ---------------------------------------------------------------------------
__global__ void final_kernel(const float* __restrict__ y_hat, const int* __restrict__ y,
                             const int* __restrict__ lp, const int* __restrict__ ln,
                             const float* __restrict__ hsum, float* __restrict__ out) {
    __shared__ float red[256];
    const int tid = threadIdx.x;
    float acc = 0.f;
    for (int i = tid; i < N; i += 256) {
        float a = y_hat[2 * i], b = y_hat[2 * i + 1];
        float m = fmaxf(a, b);
        float lse = m + logf(expf(a - m) + expf(b - m));
        float sel = (y[i] == 1) ? b : a;
        acc += sel - lse;      // logp[i, y[i]]
    }
    red[tid] = acc;
    __syncthreads();
    for (int off = 128; off > 0; off >>= 1) {
        if (tid < off) red[tid] += red[tid + off];
        __syncthreads();
    }
    if (tid == 0) {
        float class_loss = -red[0] / (float)N;
        float dloss = 0.f;
        for (int c = 0; c < KCL; ++c) {
            int d = lp[c] - 1;
            float denom = (float)(d > 1 ? d : 1);
            float lc = fmaxf(hsum[c] / denom, 0.f);
            if (lp[c] > 1 && ln[c] > 0) dloss += lc;
        }
        out[0] = ALPHA_C * class_loss + (1.0f - ALPHA_C) * dloss;
    }
}

// ---------------------------------------------------------------------------
extern "C" void kernel_launch(void* const* d_in, const int* in_sizes, int n_in,
                              void* d_out, int out_size, void* d_ws, size_t ws_size,
                              hipStream_t stream) {
    const float* x      = (const float*)d_in[0];   // [N, DIM]
    const float* y_hat  = (const float*)d_in[1];   // [N, 2]
    const int*   y      = (const int*)d_in[2];     // [N]
    const int*   labels = (const int*)d_in[3];     // [N]
    float* out = (float*)d_out;

    // workspace layout
    float* gram = (float*)d_ws;                    // N*N f32 (16 MB)
    float* sq   = gram + (size_t)N * N;            // N
    float* s    = sq + N;                          // N
    float* dpn  = s + N;                           // K*N
    float* hsum = dpn + (size_t)KCL * N;           // K
    int*   lp   = (int*)(hsum + KCL);              // K
    int*   ln   = lp + KCL;                        // K
    int*   tt   = ln + KCL;                        // K

    row_stats_kernel<<<N, 256, 0, stream>>>(x, sq, s);
    gram_wmma_kernel<<<(N / BM) * (N / BN), 256, 0, stream>>>(x, gram);
    cluster_stats_kernel<<<1, 32, 0, stream>>>(y, labels, lp, ln, tt, hsum);
    dpn_kernel<<<(KCL * N) / 256, 256, 0, stream>>>(gram, sq, s, tt, dpn);
    hinge_kernel<<<(size_t)N * N / 256, 256, 0, stream>>>(gram, sq, s, y, labels, dpn, hsum);
    final_kernel<<<1, 256, 0, stream>>>(y_hat, y, lp, ln, hsum, out);
}